// _NodeBlock_26474178413324
// MI455X (gfx1250) — compile-verified
//
#include <hip/hip_runtime.h>

#define D 128
#define D2 256
#define N_NODES 100000
#define XS 260   // LDS row stride for X (stride%64==4 -> conflict-free A-frag column reads)
#define HS 132   // LDS row stride for H buffers

#define NODES_PER_BLOCK 32          // two 16-row WMMA tiles; 100000 = 32 * 3125
#define MLP_BLOCKS (N_NODES / NODES_PER_BLOCK)

typedef __attribute__((ext_vector_type(2))) float v2f;
typedef __attribute__((ext_vector_type(8))) float v8f;

// ---------------------------------------------------------------------------
// Kernel 1: scatter-sum edge features onto destination nodes.
// One wave (32 lanes) per edge; each lane handles 4 consecutive floats.
// HBM-bound: 819 MB edge stream + fp32 hardware atomics into h_dest.
// ---------------------------------------------------------------------------
__global__ void __launch_bounds__(256)
scatter_edges_kernel(const float* __restrict__ ef, const int* __restrict__ dst,
                     float* __restrict__ hdest, int E) {
    int gid  = blockIdx.x * 256 + threadIdx.x;
    int edge = gid >> 5;
    int lane = gid & 31;
    if (edge >= E) return;
    const float4 v = reinterpret_cast<const float4*>(ef + (size_t)edge * D)[lane];
    int d = dst[edge];
    float* o = hdest + (size_t)d * D + lane * 4;
    unsafeAtomicAdd(o + 0, v.x);
    unsafeAtomicAdd(o + 1, v.y);
    unsafeAtomicAdd(o + 2, v.z);
    unsafeAtomicAdd(o + 3, v.w);
}

// ---------------------------------------------------------------------------
// One GEMM layer over a 32-node tile using V_WMMA_F32_16X16X4_F32.
// Each wave owns one 16-column tile and BOTH 16-row tiles: the B fragment
// (from global/L2) is loaded once per k-step and reused for two WMMAs,
// halving L2 dword loads per WMMA. A fragments come from LDS (b64 loads).
// ---------------------------------------------------------------------------
__device__ __forceinline__ void gemm_layer(const float* __restrict__ Xin, int xstride, int K,
                                           const float* __restrict__ W,
                                           const float* __restrict__ bias,
                                           float* __restrict__ Hout, int hstride,
                                           int n0, int lane, bool relu) {
    const int m    = lane & 15;   // A-row / B-col within tile (ISA fp32 16x4 layout)
    const int half = lane >> 4;   // lane-half selects K pair
    const int n    = n0 + m;      // output column for this lane
    v8f acc0 = {0.f, 0.f, 0.f, 0.f, 0.f, 0.f, 0.f, 0.f};
    v8f acc1 = {0.f, 0.f, 0.f, 0.f, 0.f, 0.f, 0.f, 0.f};
    for (int k0 = 0; k0 < K; k0 += 4) {
        v2f b;
        const float* bp = W + (size_t)(k0 + 2 * half) * D + n;   // rows k0+2h, k0+2h+1
        b.x = bp[0];
        b.y = bp[D];
        v2f a0, a1;
        const float* ap = Xin + m * xstride + k0 + 2 * half;
        a0.x = ap[0];
        a0.y = ap[1];
        const float* aq = ap + 16 * xstride;                     // second row-tile
        a1.x = aq[0];
        a1.y = aq[1];
        acc0 = __builtin_amdgcn_wmma_f32_16x16x4_f32(
            false, a0, false, b, (short)0, acc0, false, false);
        acc1 = __builtin_amdgcn_wmma_f32_16x16x4_f32(
            false, a1, false, b, (short)0, acc1, false, false);
    }
    const float bs = bias[n];
#pragma unroll
    for (int r = 0; r < 8; ++r) {                 // C/D: VGPR r -> row r + 8*half
        float v0 = acc0[r] + bs;
        float v1 = acc1[r] + bs;
        if (relu) {
            v0 = fmaxf(v0, 0.f);
            v1 = fmaxf(v1, 0.f);
        }
        Hout[(r + 8 * half) * hstride + n]        = v0;
        Hout[(16 + r + 8 * half) * hstride + n]   = v1;
    }
}

// ---------------------------------------------------------------------------
// Kernel 2: fused 3-layer MLP + LayerNorm + residual for a 32-node tile.
// 256 threads = 8 waves; wave w owns column tile n0 = 16*w.
// LDS aliasing: H2 reuses X's buffer (X dead after layer 1), H3 reuses H1's
// buffer (H1 dead after layer 2) -> ~52 KB LDS, high WGP occupancy.
// ---------------------------------------------------------------------------
__global__ void __launch_bounds__(256)
node_mlp_kernel(const float* __restrict__ hdest, const float* __restrict__ nodef,
                const float* __restrict__ W1, const float* __restrict__ b1,
                const float* __restrict__ W2, const float* __restrict__ b2,
                const float* __restrict__ W3, const float* __restrict__ b3,
                const float* __restrict__ gamma, const float* __restrict__ beta,
                float* __restrict__ out) {
    __shared__ float SA[NODES_PER_BLOCK * XS];        // X (32x256, stride 260); later H2
    __shared__ float SB[NODES_PER_BLOCK * HS];        // H1 (32x128, stride 132); later H3
    __shared__ float red[NODES_PER_BLOCK * 8 * 2];

    const int tid  = threadIdx.x;
    const int wave = tid >> 5;
    const int lane = tid & 31;
    const int n0   = wave * 16;                       // this wave's column tile
    const size_t node0 = (size_t)blockIdx.x * NODES_PER_BLOCK;

    // ---- stage X = concat(h_dest, node_features) into LDS (float4 traffic) ----
    {
        const int r  = tid >> 3;          // row 0..31
        const int ch = tid & 7;           // 32-float chunk 0..7
        const size_t node = node0 + r;
        const float* src = (ch < 4) ? (hdest + node * D + ch * 32)
                                    : (nodef + node * D + (ch - 4) * 32);
        float4* dl = reinterpret_cast<float4*>(SA + r * XS + ch * 32);
#pragma unroll
        for (int i = 0; i < 8; ++i)
            dl[i] = reinterpret_cast<const float4*>(src)[i];
    }
    __syncthreads();

    gemm_layer(SA, XS, D2, W1, b1, SB, HS, n0, lane, true);   // X  -> H1
    __syncthreads();
    gemm_layer(SB, HS, D,  W2, b2, SA, HS, n0, lane, true);   // H1 -> H2 (aliases X)
    __syncthreads();
    gemm_layer(SA, HS, D,  W3, b3, SB, HS, n0, lane, false);  // H2 -> H3 (aliases H1)
    __syncthreads();

    // ---- LayerNorm + gamma/beta + residual, 8 threads per row ----
    {
        const int r  = tid >> 3;          // row 0..31
        const int ch = tid & 7;
        float s = 0.f, s2 = 0.f;
#pragma unroll
        for (int j = 0; j < 16; ++j) {
            float x = SB[r * HS + ch * 16 + j];
            s  += x;
            s2 += x * x;
        }
        red[(r * 8 + ch) * 2 + 0] = s;
        red[(r * 8 + ch) * 2 + 1] = s2;
        __syncthreads();
        float S = 0.f, S2 = 0.f;
#pragma unroll
        for (int j = 0; j < 8; ++j) {
            S  += red[(r * 8 + j) * 2 + 0];
            S2 += red[(r * 8 + j) * 2 + 1];
        }
        const float mu   = S * (1.0f / D);
        const float var  = S2 * (1.0f / D) - mu * mu;
        const float rsig = rsqrtf(var + 1e-5f);
        const size_t node = node0 + r;
#pragma unroll
        for (int j = 0; j < 16; ++j) {
            const int c = ch * 16 + j;
            float x = SB[r * HS + c];
            out[node * D + c] =
                (x - mu) * rsig * gamma[c] + beta[c] + nodef[node * D + c];
        }
    }
}

// ---------------------------------------------------------------------------
extern "C" void kernel_launch(void* const* d_in, const int* in_sizes, int n_in,
                              void* d_out, int out_size, void* d_ws, size_t ws_size,
                              hipStream_t stream) {
    const float* nodef = (const float*)d_in[0];
    const float* ef    = (const float*)d_in[1];
    const int*   dst   = (const int*)  d_in[2];
    const float* W1    = (const float*)d_in[3];
    const float* b1    = (const float*)d_in[4];
    const float* W2    = (const float*)d_in[5];
    const float* b2    = (const float*)d_in[6];
    const float* W3    = (const float*)d_in[7];
    const float* b3    = (const float*)d_in[8];
    const float* gamma = (const float*)d_in[9];
    const float* beta  = (const float*)d_in[10];
    float* out   = (float*)d_out;
    float* hdest = (float*)d_ws;                     // [N_NODES x D] scatter accumulator

    const int E = in_sizes[1] / D;

    hipMemsetAsync(hdest, 0, (size_t)N_NODES * D * sizeof(float), stream);

    // one wave per edge -> E*32 threads
    const int scatter_blocks = (E * 32 + 255) / 256;
    scatter_edges_kernel<<<scatter_blocks, 256, 0, stream>>>(ef, dst, hdest, E);

    node_mlp_kernel<<<MLP_BLOCKS, 256, 0, stream>>>(
        hdest, nodef, W1, b1, W2, b2, W3, b3, gamma, beta, out);
}